// BackProjectionLayer_88691074662810
// MI455X (gfx1250) — compile-verified
//
#include <hip/hip_runtime.h>
#include <math.h>

// Problem constants (match reference)
#define A_N   180
#define B_N   2
#define HR_N  512
#define HW_O  512
#define PAD   51
#define HP    614              // HW_O + 2*PAD
#define WP    614
#define NCOL  (A_N * B_N)      // 360 (a,b) columns
#define NGRP  ((HP + 15) / 16) // 39 row groups of hp
#define NTIL  ((NCOL + 15)/16) // 23 column tiles

typedef __attribute__((ext_vector_type(2))) float v2f;
typedef __attribute__((ext_vector_type(8))) float v8f;

// ---------------------------------------------------------------------------
// Kernel 1: build col[a,b,hp] = linear-resampled sinogram rows (512 -> 614)
// as a banded 16x16 GEMM per tile using V_WMMA_F32_16X16X4_F32.
// One wave (32 threads) per (row-group g, column-tile j). Body is branch-free
// (EXEC all-ones for WMMA); out-of-range columns/rows are clamped for loads
// and masked only at the store.
// ---------------------------------------------------------------------------
__global__ __launch_bounds__(32) void col_wmma_kernel(
    const float* __restrict__ sino,   // (B, A, HR)
    float* __restrict__ colt)         // (A*B, HP)  index p*HP + hp, p = a*2+b
{
  const int g    = blockIdx.x;        // 0..NGRP-1
  const int j    = blockIdx.y;        // 0..NTIL-1
  const int lane = threadIdx.x;       // 0..31 (wave32)
  const int half = lane >> 4;

  const float scale = (float)HR_N / (float)HP;

  // Base sino bin for this 16-row group (band start), clamped so the
  // 16-wide window stays inside [0, HR).
  float src0 = (16.0f * (float)g + 0.5f) * scale - 0.5f;
  src0 = fminf(fmaxf(src0, 0.0f), (float)(HR_N - 1));
  int base = (int)floorf(src0);
  if (base > HR_N - 16) base = HR_N - 16;

  // This lane's A-matrix row: hp = 16*g + row -> taps i0/i1, weight w.
  // Rows beyond HP are clamped; their results are masked at the store.
  const int hp = min(16 * g + (lane & 15), HP - 1);
  float srcr = ((float)hp + 0.5f) * scale - 0.5f;
  srcr = fminf(fmaxf(srcr, 0.0f), (float)(HR_N - 1));
  const int   i0 = (int)floorf(srcr);
  const float w  = srcr - (float)i0;
  const int   i1 = min(i0 + 1, HR_N - 1);

  // This lane's B-matrix column: n -> (a,b) pair p. Clamp for loads,
  // mask at store.
  const int  p   = j * 16 + (lane & 15);
  const bool pok = (p < NCOL);
  const int  ai  = min(p >> 1, A_N - 1);
  const int  bi  = p & 1;
  const float* srow = sino + ((size_t)bi * A_N + (size_t)ai) * HR_N;

  v8f acc = {};
#pragma unroll
  for (int s = 0; s < 4; ++s) {
    // K-slice s covers absolute bins base+4s .. base+4s+3.
    // A 16x4 layout: lanes 0-15 hold K=0(v0),1(v1); lanes 16-31 hold K=2,3.
    const int k0 = base + 4 * s + (half ? 2 : 0);
    const int k1 = k0 + 1;

    v2f afrag;
    afrag.x = ((k0 == i0) ? (1.0f - w) : 0.0f) + ((k0 == i1) ? w : 0.0f);
    afrag.y = ((k1 == i0) ? (1.0f - w) : 0.0f) + ((k1 == i1) ? w : 0.0f);

    // B 4x16 layout mirrors A: lane holds column n; halves split K.
    v2f bfrag;
    bfrag.x = srow[k0];
    bfrag.y = srow[k1];

    acc = __builtin_amdgcn_wmma_f32_16x16x4_f32(
        /*neg_a=*/false, afrag, /*neg_b=*/false, bfrag,
        /*c_mod=*/(short)0, acc, /*reuse_a=*/false, /*reuse_b=*/false);
  }

  // D 16x16 layout: VGPR r -> M = r (lanes 0-15) / r+8 (lanes 16-31), N = lane%16.
#pragma unroll
  for (int r = 0; r < 8; ++r) {
    const int M   = r + (half ? 8 : 0);
    const int hpo = 16 * g + M;
    if (pok && hpo < HP) colt[(size_t)p * HP + hpo] = acc[r];
  }
}

// ---------------------------------------------------------------------------
// Kernel 2: backprojection. One thread per output pixel; 180-angle loop with
// 2-tap gathers from the L2-resident col table. Trig staged in LDS per block.
// ---------------------------------------------------------------------------
__global__ __launch_bounds__(256) void backproj_kernel(
    const float* __restrict__ colt,     // (A*B, HP)
    const float* __restrict__ angles,   // (A,) degrees
    float* __restrict__ out)            // (B,1,H,W) flat
{
  __shared__ float s_cos[A_N];
  __shared__ float s_sin[A_N];
  const int t = threadIdx.x;
  if (t < A_N) {
    const float th = -angles[t] * 0.017453292519943295f;  // -deg2rad
    float sn, cs;
    sincosf(th, &sn, &cs);
    s_cos[t] = cs;
    s_sin[t] = sn;
  }
  __syncthreads();

  const int gid = blockIdx.x * 256 + t;            // < 2*512*512
  const int x = gid & (HW_O - 1);
  const int y = (gid >> 9) & (HW_O - 1);
  const int b = gid >> 18;

  const float xs = ((float)(x + PAD) + 0.5f) * (2.0f / (float)WP) - 1.0f;
  const float ys = ((float)(y + PAD) + 0.5f) * (2.0f / (float)HP) - 1.0f;

  const float* cp = colt + (size_t)b * HP;         // advance by B_N*HP per angle
  float acc = 0.0f;
  for (int a = 0; a < A_N; ++a, cp += (size_t)B_N * HP) {
    const float c = s_cos[a];
    const float s = s_sin[a];
    const float gx = c * xs - s * ys;
    const float gy = s * xs + c * ys;
    // ((g+1)*614 - 1)*0.5 = g*307 + 306.5
    const float ix = gx * 307.0f + 306.5f;
    const float iy = gy * 307.0f + 306.5f;

    const float x0f = floorf(ix);
    const float wx  = ix - x0f;
    const int   x0  = (int)x0f;
    float fx = 0.0f;
    if (x0 >= 0     && x0     < WP) fx += 1.0f - wx;
    if (x0 + 1 >= 0 && x0 + 1 < WP) fx += wx;

    const float y0f = floorf(iy);
    const float wy  = iy - y0f;
    const int   y0  = (int)y0f;
    const int   y1  = y0 + 1;
    const int   y0c = min(max(y0, 0), HP - 1);
    const int   y1c = min(max(y1, 0), HP - 1);
    const float fy0 = (y0 >= 0 && y0 < HP) ? (1.0f - wy) : 0.0f;
    const float fy1 = (y1 >= 0 && y1 < HP) ? wy : 0.0f;

    acc += fx * (fy0 * cp[y0c] + fy1 * cp[y1c]);
  }
  out[gid] = acc * (1.0f / (180.0f + 1e-6f));
}

// ---------------------------------------------------------------------------
// Fallback (only if d_ws cannot hold the col table): fuse the 2-tap detector
// interpolation into the backprojection loop, reading sino directly.
// ---------------------------------------------------------------------------
__global__ __launch_bounds__(256) void backproj_direct_kernel(
    const float* __restrict__ sino,
    const float* __restrict__ angles,
    float* __restrict__ out)
{
  __shared__ float s_cos[A_N];
  __shared__ float s_sin[A_N];
  const int t = threadIdx.x;
  if (t < A_N) {
    const float th = -angles[t] * 0.017453292519943295f;
    float sn, cs;
    sincosf(th, &sn, &cs);
    s_cos[t] = cs;
    s_sin[t] = sn;
  }
  __syncthreads();

  const int gid = blockIdx.x * 256 + t;
  const int x = gid & (HW_O - 1);
  const int y = (gid >> 9) & (HW_O - 1);
  const int b = gid >> 18;

  const float xs = ((float)(x + PAD) + 0.5f) * (2.0f / (float)WP) - 1.0f;
  const float ys = ((float)(y + PAD) + 0.5f) * (2.0f / (float)HP) - 1.0f;
  const float scale = (float)HR_N / (float)HP;

  float acc = 0.0f;
  for (int a = 0; a < A_N; ++a) {
    const float c = s_cos[a];
    const float s = s_sin[a];
    const float gx = c * xs - s * ys;
    const float gy = s * xs + c * ys;
    const float ix = gx * 307.0f + 306.5f;
    const float iy = gy * 307.0f + 306.5f;

    const float x0f = floorf(ix);
    const float wx  = ix - x0f;
    const int   x0  = (int)x0f;
    float fx = 0.0f;
    if (x0 >= 0     && x0     < WP) fx += 1.0f - wx;
    if (x0 + 1 >= 0 && x0 + 1 < WP) fx += wx;

    const float y0f = floorf(iy);
    const float wy  = iy - y0f;
    const int   y0  = (int)y0f;
    const int   y1  = y0 + 1;
    const int   y0c = min(max(y0, 0), HP - 1);
    const int   y1c = min(max(y1, 0), HP - 1);
    const float fy0 = (y0 >= 0 && y0 < HP) ? (1.0f - wy) : 0.0f;
    const float fy1 = (y1 >= 0 && y1 < HP) ? wy : 0.0f;

    const float* sp = sino + ((size_t)b * A_N + a) * HR_N;
    float v0, v1;
    {
      float sr = fminf(fmaxf(((float)y0c + 0.5f) * scale - 0.5f, 0.0f), (float)(HR_N - 1));
      int ii0 = (int)floorf(sr);
      float ww = sr - (float)ii0;
      int ii1 = min(ii0 + 1, HR_N - 1);
      v0 = sp[ii0] * (1.0f - ww) + sp[ii1] * ww;
    }
    {
      float sr = fminf(fmaxf(((float)y1c + 0.5f) * scale - 0.5f, 0.0f), (float)(HR_N - 1));
      int ii0 = (int)floorf(sr);
      float ww = sr - (float)ii0;
      int ii1 = min(ii0 + 1, HR_N - 1);
      v1 = sp[ii0] * (1.0f - ww) + sp[ii1] * ww;
    }
    acc += fx * (fy0 * v0 + fy1 * v1);
  }
  out[gid] = acc * (1.0f / (180.0f + 1e-6f));
}

extern "C" void kernel_launch(void* const* d_in, const int* in_sizes, int n_in,
                              void* d_out, int out_size, void* d_ws, size_t ws_size,
                              hipStream_t stream) {
  const float* sino   = (const float*)d_in[0];   // (2,180,512) f32
  const float* angles = (const float*)d_in[1];   // (180,) f32
  float* out = (float*)d_out;                    // (2,1,512,512) f32

  const size_t colbytes = (size_t)NCOL * HP * sizeof(float);  // ~884 KB
  const int nblocks = (B_N * HW_O * HW_O) / 256;              // 2048

  if (ws_size >= colbytes) {
    float* colt = (float*)d_ws;
    dim3 g1(NGRP, NTIL);                                      // 39 x 23 waves
    col_wmma_kernel<<<g1, 32, 0, stream>>>(sino, colt);
    backproj_kernel<<<nblocks, 256, 0, stream>>>(colt, angles, out);
  } else {
    backproj_direct_kernel<<<nblocks, 256, 0, stream>>>(sino, angles, out);
  }
}